// MoE_19825569038534
// MI455X (gfx1250) — compile-verified
//
#include <hip/hip_runtime.h>
#include <math.h>

typedef float v2f __attribute__((ext_vector_type(2)));
typedef float v4f __attribute__((ext_vector_type(4)));
typedef float v8f __attribute__((ext_vector_type(8)));

#define LOG100 4.605170185988092f

// ---------------------------------------------------------------------------
// wave32 butterfly all-reduce
// ---------------------------------------------------------------------------
__device__ __forceinline__ float wave_sum(float v) {
#pragma unroll
  for (int m = 16; m >= 1; m >>= 1) v += __shfl_xor(v, m, 32);
  return v;
}

// ---------------------------------------------------------------------------
// GEMM tiling: block = 256 threads (8 waves), tile 64(M) x 128(N), BK=32.
// Waves arranged 4(M) x 2(N); each wave computes 16x64 = four 16x16 f32
// accumulators via V_WMMA_F32_16X16X4_F32.
//
// LDS layouts:
//   As[row][k]            : row-major, fragment = contiguous float2 (b64)
//   Bs[kpair][n*2 + 0/1]  : k-pair interleaved so fragment {B[k][n],B[k+1][n]}
//                           is one contiguous float2 (b64)
// ---------------------------------------------------------------------------
#define BM 64
#define BN 128
#define BK 32
#define ASTRIDE (BK + 4)
#define BPSTRIDE (2 * BN + 8)

// ---- staging helpers (256 threads) ----------------------------------------
__device__ __forceinline__ void stage_A(float* As, const float* __restrict__ A,
                                        int m0, int K, int k0, int tid) {
#pragma unroll
  for (int j = 0; j < 2; ++j) {
    int idx = tid * 2 + j;  // 0..511
    int r   = idx >> 3;     // row 0..63
    int c4  = (idx & 7) * 4;
    const float* src = A + (size_t)(m0 + r) * K + (k0 + c4);
    v4f v = *(const v4f*)src;
    *(v4f*)&As[r * ASTRIDE + c4] = v;
    __builtin_prefetch(src + BK, 0, 1);  // next K-chunk
  }
}

__device__ __forceinline__ void stage_B(float* Bs, const float* __restrict__ W,
                                        int n0, int Nd, int k0, int tid,
                                        size_t pf_delta) {
#pragma unroll
  for (int j = 0; j < 2; ++j) {
    int idx = tid * 2 + j;   // 0..511
    int pr  = idx >> 5;      // k-pair 0..15
    int c4  = (idx & 31) * 4;
    const float* src = W + (size_t)(k0 + 2 * pr) * Nd + (n0 + c4);
    v4f r0 = *(const v4f*)src;
    v4f r1 = *(const v4f*)(src + Nd);
    v4f lo = {r0.x, r1.x, r0.y, r1.y};
    v4f hi = {r0.z, r1.z, r0.w, r1.w};
    float* dst = &Bs[pr * BPSTRIDE + c4 * 2];
    *(v4f*)dst       = lo;
    *(v4f*)(dst + 4) = hi;
    __builtin_prefetch(src + pf_delta, 0, 1);
  }
}

// C[M,Nd] = A[M,K] @ W[K,Nd] + bias[Nd]   (router projection)
__global__ __launch_bounds__(256) void gemm_bias_kernel(
    const float* __restrict__ A, const float* __restrict__ W,
    const float* __restrict__ bias, float* __restrict__ C,
    int M, int K, int Nd) {
  __shared__ float As[BM * ASTRIDE];
  __shared__ float Bs[(BK / 2) * BPSTRIDE];

  const int tid  = threadIdx.x;
  const int lane = tid & 31;
  const int wave = tid >> 5;
  const int mw   = wave >> 1;   // 0..3
  const int nw   = wave & 1;    // 0..1
  const int m0   = blockIdx.y * BM;
  const int n0   = blockIdx.x * BN;
  const int l15  = lane & 15;
  const int hi   = lane >> 4;

  v8f acc[4];
#pragma unroll
  for (int t = 0; t < 4; ++t) acc[t] = (v8f){};

  for (int k0 = 0; k0 < K; k0 += BK) {
    if (k0) __syncthreads();  // previous chunk's readers done
    stage_A(As, A, m0, K, k0, tid);
    stage_B(Bs, W, n0, Nd, k0, tid, (size_t)BK * Nd);
    __syncthreads();

    v2f af[8];
#pragma unroll
    for (int q = 0; q < 8; ++q)
      af[q] = *(const v2f*)&As[(mw * 16 + l15) * ASTRIDE + q * 4 + 2 * hi];

#pragma unroll
    for (int q = 0; q < 8; ++q) {
#pragma unroll
      for (int t = 0; t < 4; ++t) {
        v2f b = *(const v2f*)&Bs[(q * 2 + hi) * BPSTRIDE +
                                 (nw * 64 + t * 16 + l15) * 2];
        acc[t] = __builtin_amdgcn_wmma_f32_16x16x4_f32(
            false, af[q], false, b, (short)0, acc[t], false, false);
      }
    }
  }

#pragma unroll
  for (int t = 0; t < 4; ++t) {
    int col  = n0 + nw * 64 + t * 16 + l15;
    float bv = bias[col];
#pragma unroll
    for (int r = 0; r < 8; ++r) {
      int row = m0 + mw * 16 + r + 8 * hi;
      C[(size_t)row * Nd + col] = acc[t][r] + bv;
    }
  }
}

// out = exp(min(lt,log100)) * [ sum_e (diag(gate_e) H) @ W_e + gate_e*b_e ]
// Gate scales rows of A, which is a per-lane scalar on the A fragment, so all
// experts and all K accumulate into one accumulator set.
__global__ __launch_bounds__(256) void moe_expert_kernel(
    const float* __restrict__ H, const float* __restrict__ EW,
    const float* __restrict__ Eb, const float* __restrict__ gate,
    const float* __restrict__ layer_temp, float* __restrict__ out,
    int M, int K, int Nd) {
  __shared__ float As[BM * ASTRIDE];
  __shared__ float Bs[(BK / 2) * BPSTRIDE];

  const int tid  = threadIdx.x;
  const int lane = tid & 31;
  const int wave = tid >> 5;
  const int mw   = wave >> 1;
  const int nw   = wave & 1;
  const int m0   = blockIdx.y * BM;
  const int n0   = blockIdx.x * BN;
  const int l15  = lane & 15;
  const int hi   = lane >> 4;

  // per-lane gates for the A-fragment row m = m0 + mw*16 + l15
  const float* grow = gate + (size_t)(m0 + mw * 16 + l15) * 8;
  v4f g0 = *(const v4f*)grow;
  v4f g1 = *(const v4f*)(grow + 4);
  float g[8] = {g0.x, g0.y, g0.z, g0.w, g1.x, g1.y, g1.z, g1.w};

  v8f acc[4];
#pragma unroll
  for (int t = 0; t < 4; ++t) acc[t] = (v8f){};

  for (int k0 = 0; k0 < K; k0 += BK) {
    if (k0) __syncthreads();
    stage_A(As, H, m0, K, k0, tid);  // staged once per K-chunk (not per expert)

    v2f af[8];
#pragma unroll
    for (int e = 0; e < 8; ++e) {
      if (e) __syncthreads();  // previous expert's B readers done
      stage_B(Bs, EW + (size_t)e * K * Nd, n0, Nd, k0, tid,
              (size_t)K * Nd);  // prefetch next expert's tile
      __syncthreads();
      if (e == 0) {
#pragma unroll
        for (int q = 0; q < 8; ++q)
          af[q] = *(const v2f*)&As[(mw * 16 + l15) * ASTRIDE + q * 4 + 2 * hi];
      }
      float ge = g[e];
#pragma unroll
      for (int q = 0; q < 8; ++q) {
        v2f ag;
        ag.x = af[q].x * ge;
        ag.y = af[q].y * ge;
#pragma unroll
        for (int t = 0; t < 4; ++t) {
          v2f b = *(const v2f*)&Bs[(q * 2 + hi) * BPSTRIDE +
                                   (nw * 64 + t * 16 + l15) * 2];
          acc[t] = __builtin_amdgcn_wmma_f32_16x16x4_f32(
              false, ag, false, b, (short)0, acc[t], false, false);
        }
      }
    }
  }

  // epilogue: + sum_e gate[row][e]*b_e[col], then * exp(min(lt,log100))
  float ls = expf(fminf(layer_temp[0], LOG100));
  float ebv[4][8];
#pragma unroll
  for (int t = 0; t < 4; ++t) {
    int col = n0 + nw * 64 + t * 16 + l15;
#pragma unroll
    for (int e = 0; e < 8; ++e) ebv[t][e] = Eb[(size_t)e * Nd + col];
  }
#pragma unroll
  for (int r = 0; r < 8; ++r) {
    int row = m0 + mw * 16 + r + 8 * hi;
    const float* gr = gate + (size_t)row * 8;
    v4f ga = *(const v4f*)gr;
    v4f gb = *(const v4f*)(gr + 4);
    float grw[8] = {ga.x, ga.y, ga.z, ga.w, gb.x, gb.y, gb.z, gb.w};
#pragma unroll
    for (int t = 0; t < 4; ++t) {
      int col = n0 + nw * 64 + t * 16 + l15;
      float bsum = 0.f;
#pragma unroll
      for (int e = 0; e < 8; ++e) bsum += grw[e] * ebv[t][e];
      out[(size_t)row * Nd + col] = (acc[t][r] + bsum) * ls;
    }
  }
}

// ---------------------------------------------------------------------------
// normalize sim columns: snorm[k][e] = sim[k][e] / max(||sim[:,e]||, 1e-12)
// launch: 1 block, 256 threads (wave e handles column e)
// ---------------------------------------------------------------------------
__global__ void colnorm_kernel(const float* __restrict__ sim,
                               float* __restrict__ snorm) {
  int e    = threadIdx.x >> 5;
  int lane = threadIdx.x & 31;
  float ss = 0.f;
  for (int k = lane; k < 1024; k += 32) {
    float v = sim[k * 8 + e];
    ss += v * v;
  }
  ss = wave_sum(ss);
  float inv = 1.0f / fmaxf(sqrtf(ss), 1e-12f);
  for (int k = lane; k < 1024; k += 32)
    snorm[k * 8 + e] = sim[k * 8 + e] * inv;
}

// ---------------------------------------------------------------------------
// router: wave per token. zraw row -> normalize -> logits vs snorm -> top-4
// mask, softmax, gate; atomic importance/load stats.
// ---------------------------------------------------------------------------
__global__ __launch_bounds__(256) void router_kernel(
    const float* __restrict__ zraw, const float* __restrict__ snorm,
    const float* __restrict__ rtemp, float* __restrict__ gate,
    float* __restrict__ imp, float* __restrict__ loadv, int M) {
  int wave = threadIdx.x >> 5;
  int lane = threadIdx.x & 31;
  int tok  = blockIdx.x * 8 + wave;
  if (tok >= M) return;

  const float* zr = zraw + (size_t)tok * 1024;
  float sumsq = 0.f;
  float p[8];
#pragma unroll
  for (int e = 0; e < 8; ++e) p[e] = 0.f;

  for (int k = lane; k < 1024; k += 32) {
    float z = zr[k];
    sumsq += z * z;
    const float* sn = snorm + k * 8;
    v4f s0 = *(const v4f*)sn;
    v4f s1 = *(const v4f*)(sn + 4);
    p[0] += z * s0.x; p[1] += z * s0.y; p[2] += z * s0.z; p[3] += z * s0.w;
    p[4] += z * s1.x; p[5] += z * s1.y; p[6] += z * s1.z; p[7] += z * s1.w;
  }
  sumsq = wave_sum(sumsq);
#pragma unroll
  for (int e = 0; e < 8; ++e) p[e] = wave_sum(p[e]);

  float tscale = expf(fminf(rtemp[0], LOG100));
  float inv    = tscale / fmaxf(sqrtf(sumsq), 1e-12f);
  float logits[8];
#pragma unroll
  for (int e = 0; e < 8; ++e) logits[e] = p[e] * inv;

  // top-4 mask (first-occurrence argmax matches lax.top_k tie behavior)
  float mask[8], tmp[8];
#pragma unroll
  for (int e = 0; e < 8; ++e) { mask[e] = 0.f; tmp[e] = logits[e]; }
#pragma unroll
  for (int i = 0; i < 4; ++i) {
    int   am = 0;
    float mv = tmp[0];
#pragma unroll
    for (int e = 1; e < 8; ++e)
      if (tmp[e] > mv) { mv = tmp[e]; am = e; }
    mask[am] = 1.f;
    tmp[am]  = -INFINITY;
  }

  // softmax over all 8
  float mx = logits[0];
#pragma unroll
  for (int e = 1; e < 8; ++e) mx = fmaxf(mx, logits[e]);
  float se = 0.f, pr[8];
#pragma unroll
  for (int e = 0; e < 8; ++e) { pr[e] = expf(logits[e] - mx); se += pr[e]; }
  float rse = 1.0f / se;
#pragma unroll
  for (int e = 0; e < 8; ++e) pr[e] *= rse;

  if (lane == 0) {
#pragma unroll
    for (int e = 0; e < 8; ++e) gate[(size_t)tok * 8 + e] = mask[e] * pr[e];
#pragma unroll
    for (int e = 0; e < 8; ++e) {
      atomicAdd(&imp[e], pr[e]);
      atomicAdd(&loadv[e], mask[e]);
    }
  }
}

// ---------------------------------------------------------------------------
// LayerNorm (biased var) + ReLU, wave per token, D=1024
// ---------------------------------------------------------------------------
__global__ __launch_bounds__(256) void layernorm_relu_kernel(
    const float* __restrict__ h, const float* __restrict__ scale,
    const float* __restrict__ bias, float* __restrict__ out, int M) {
  int wave = threadIdx.x >> 5;
  int lane = threadIdx.x & 31;
  int tok  = blockIdx.x * 8 + wave;
  if (tok >= M) return;
  const float* hr = h + (size_t)tok * 1024;
  float s = 0.f, ss = 0.f;
  for (int k = lane; k < 1024; k += 32) {
    float v = hr[k];
    s += v;
    ss += v * v;
  }
  s  = wave_sum(s);
  ss = wave_sum(ss);
  float mean = s * (1.0f / 1024.0f);
  float var  = ss * (1.0f / 1024.0f) - mean * mean;
  float rstd = rsqrtf(var + 1e-5f);
  float* o = out + (size_t)tok * 1024;
  for (int k = lane; k < 1024; k += 32) {
    float v = (hr[k] - mean) * rstd * scale[k] + bias[k];
    o[k] = fmaxf(v, 0.f);
  }
}

// ---------------------------------------------------------------------------
// aux loss: 0.01 * (cv^2(importance) + cv^2(load)) summed over 3 layers
// stats layout: [layer][ imp(8) | load(8) ]
// ---------------------------------------------------------------------------
__global__ void aux_kernel(const float* __restrict__ stats,
                           float* __restrict__ out) {
  if (threadIdx.x != 0 || blockIdx.x != 0) return;
  float aux = 0.f;
  for (int l = 0; l < 3; ++l) {
    for (int part = 0; part < 2; ++part) {
      const float* v = stats + l * 16 + part * 8;
      float s = 0.f;
      for (int e = 0; e < 8; ++e) s += v[e];
      float mean = s * (1.0f / 8.0f);
      float var = 0.f;
      for (int e = 0; e < 8; ++e) {
        float d = v[e] - mean;
        var += d * d;
      }
      var *= (1.0f / 7.0f);  // ddof=1
      aux += 0.01f * var / (mean * mean + 1e-10f);
    }
  }
  out[0] = aux;
}

__global__ void zero_kernel(float* __restrict__ p, int n) {
  int i = blockIdx.x * blockDim.x + threadIdx.x;
  if (i < n) p[i] = 0.f;
}

// ---------------------------------------------------------------------------
// orchestration
// ---------------------------------------------------------------------------
extern "C" void kernel_launch(void* const* d_in, const int* in_sizes, int n_in,
                              void* d_out, int out_size, void* d_ws,
                              size_t ws_size, hipStream_t stream) {
  (void)in_sizes; (void)n_in; (void)out_size; (void)ws_size;
  const int M = 8192;
  const int DI[3] = {512, 1024, 1024};
  const int DO[3] = {1024, 1024, 512};

  const float* x = (const float*)d_in[0];
  // per-layer params: base = 1 + 7*l
  const float *projW[3], *projb[3], *sim[3], *rtemp[3], *eW[3], *eb[3], *lt[3];
  for (int l = 0; l < 3; ++l) {
    int b = 1 + 7 * l;
    projW[l] = (const float*)d_in[b + 0];
    projb[l] = (const float*)d_in[b + 1];
    sim[l]   = (const float*)d_in[b + 2];
    rtemp[l] = (const float*)d_in[b + 3];
    eW[l]    = (const float*)d_in[b + 4];
    eb[l]    = (const float*)d_in[b + 5];
    lt[l]    = (const float*)d_in[b + 6];
  }
  const float* lnS[2] = {(const float*)d_in[22], (const float*)d_in[24]};
  const float* lnB[2] = {(const float*)d_in[23], (const float*)d_in[25]};

  float* ws    = (float*)d_ws;
  float* buf0  = ws;                       // 8192*1024 (zraw / layer1 out)
  float* buf1  = ws + 8388608;             // 8192*1024 (post-LN activations)
  float* gate  = ws + 16777216;            // 8192*8
  float* snorm = gate + 65536;             // 1024*8
  float* stats = snorm + 8192;             // 3*16

  float* h_emb   = (float*)d_out;            // [8192,1024] layer-0 output
  float* h_final = (float*)d_out + 8388608;  // [8192,512]
  float* aux_out = (float*)d_out + 12582912; // scalar

  zero_kernel<<<1, 64, 0, stream>>>(stats, 48);

  const float* hin[3]  = {x, buf1, buf1};
  float*       hout[3] = {h_emb, buf0, h_final};

  for (int l = 0; l < 3; ++l) {
    int K = DI[l], Nd = DO[l];
    colnorm_kernel<<<1, 256, 0, stream>>>(sim[l], snorm);
    // router projection: zraw = h @ proj_W + proj_b  -> buf0
    gemm_bias_kernel<<<dim3(1024 / BN, M / BM), 256, 0, stream>>>(
        hin[l], projW[l], projb[l], buf0, M, K, 1024);
    router_kernel<<<M / 8, 256, 0, stream>>>(
        buf0, snorm, rtemp[l], gate, stats + l * 16, stats + l * 16 + 8, M);
    // experts (layer-1 output overwrites buf0; zraw is dead once gate exists)
    moe_expert_kernel<<<dim3(Nd / BN, M / BM), 256, 0, stream>>>(
        hin[l], eW[l], eb[l], gate, lt[l], hout[l], M, K, Nd);
    if (l < 2) {
      layernorm_relu_kernel<<<M / 8, 256, 0, stream>>>(
          hout[l], lnS[l], lnB[l], buf1, M);
    }
  }
  aux_kernel<<<1, 32, 0, stream>>>(stats, aux_out);
}